// WKV_50379966382716
// MI455X (gfx1250) — compile-verified
//
#include <hip/hip_runtime.h>

// CDNA5 / gfx1250 WKV (RWKV matrix-state) scan, chunked linear-attention form.
// f32 WMMA 16x16x4 path; state column-partitioned across 8 workgroups,
// resident in LDS (320KB/WGP on CDNA5); v-stripe staged by the Tensor Data
// Mover (TDM) with double buffering. wave32.

typedef float v2f __attribute__((ext_vector_type(2)));
typedef float v8f __attribute__((ext_vector_type(8)));
typedef unsigned int v4u __attribute__((ext_vector_type(4)));
typedef int v4i __attribute__((ext_vector_type(4)));
typedef int v8i __attribute__((ext_vector_type(8)));

#define DIMD   512
#define TT     2048
#define CC     16            // chunk length (one 16-row WMMA tile of timesteps)
#define BCOL   64            // state columns owned per workgroup
#define SP     68            // padded LDS row stride for state (bank-conflict pad)
#define NCHUNK (TT / CC)
#define NTHR   128           // 4 waves of 32

// Issue a TDM 2D-tile load: 64x16 f32 tile of the (2048 x 512) v tensor -> LDS.
// D# packing per CDNA5 ISA (group0 128b, group1 256b).
__device__ __forceinline__ void tdm_load_v_tile(const float* gsrc, unsigned lds_off) {
  unsigned long long ga = (unsigned long long)(uintptr_t)gsrc;
  v4u g0;
  g0[0] = 1u;                                     // count=1, user desc
  g0[1] = lds_off;                                // lds_addr (bytes)
  g0[2] = (unsigned)(ga & 0xFFFFFFFFu);           // global_addr[31:0]
  g0[3] = (unsigned)((ga >> 32) & 0x01FFFFFFu)    // global_addr[56:32]
          | (2u << 30);                           // type=2 ("image")
  v8i g1;
  g1[0] = 0x20000;                                // wg_mask=0, data_size=2 (4B)
  g1[1] = (int)(512u << 16);                      // tensor_dim0 = 512 (lo16)
  g1[2] = (int)(2048u << 16);                     // dim0 hi=0 | tensor_dim1 = 2048 (lo16)
  g1[3] = (int)(64u << 16);                       // dim1 hi=0 | tile_dim0 = 64
  g1[4] = 16;                                     // tile_dim1 = 16, tile_dim2 = 0
  g1[5] = 512;                                    // tensor_dim0_stride = 512 (lo32)
  g1[6] = 0;                                      // stride hi, dim1_stride lo
  g1[7] = 0;
  v4i z4 = {0, 0, 0, 0};
#if defined(__clang_major__) && (__clang_major__ >= 23)
  v8i z8 = {0, 0, 0, 0, 0, 0, 0, 0};
  __builtin_amdgcn_tensor_load_to_lds(g0, g1, z4, z4, z8, 0);
#else
  __builtin_amdgcn_tensor_load_to_lds(g0, g1, z4, z4, 0);
#endif
}

__global__ __launch_bounds__(NTHR) void wkv_cdna5_kernel(
    const float* __restrict__ r, const float* __restrict__ w,
    const float* __restrict__ k, const float* __restrict__ v,
    const float* __restrict__ init_state, const float* __restrict__ u,
    float* __restrict__ out)
{
  extern __shared__ float smem[];
  float* st   = smem;                 // [512][SP]   state block (rows a, local cols)
  float* rt   = st + DIMD * SP;       // [16][512]   r~ = r * P_tau (exclusive cumdecay)
  float* kt   = rt + CC * DIMD;       // [16][512]   k~ = k / P_{s+1}
  float* vt   = kt + CC * DIMD;       // [2][16][64] v stripe, TDM double buffer
  float* Pt   = vt + 2 * CC * BCOL;   // [512]       total chunk decay P_C per row
  float* sc   = Pt + DIMD;            // [16][16]    intra-chunk scores (masked)
  float* dtmp = sc + CC * CC;         // [16]        diagonal u-term accumulators

  const int tid    = threadIdx.x;
  const int lane   = tid & 31;
  const int wid    = tid >> 5;        // wave id 0..3, owns 16-col tile
  const int hi     = lane >> 4;       // lane half (WMMA K/M-half select)
  const int ln     = lane & 15;       // M (A-frag) or N (B/C-frag) index
  const int wgBase = blockIdx.x * BCOL;
  const int n0     = wid * 16;        // this wave's column-tile offset in block

  const unsigned ldsBase = __builtin_amdgcn_groupstaticsize();
  const unsigned vtOff0  = ldsBase + (unsigned)((char*)vt - (char*)smem);

  // state0[a][b] = init_state[b]  (broadcast along rows)
  for (int i = tid; i < DIMD * BCOL; i += NTHR) {
    int a = i >> 6, b = i & 63;
    st[a * SP + b] = init_state[wgBase + b];
  }

  // Prime TDM pipeline: chunk 0's v tile -> buffer 0 (one wave; EXEC ignored).
  if (wid == 3)
    tdm_load_v_tile(v + (size_t)0 * CC * DIMD + wgBase, vtOff0);

  float* yout = out;                          // (T, 512)
  float* sout = out + (size_t)TT * DIMD;      // (512, 512) final state

  for (int c0 = 0; c0 < NCHUNK; ++c0) {
    if (tid < CC) dtmp[tid] = 0.0f;
    __syncthreads();  // also fences previous chunk's rt/kt/st/vt readers

    // Kick off next chunk's v tile into the other buffer (overlaps all compute).
    if (wid == 3 && c0 + 1 < NCHUNK)
      tdm_load_v_tile(v + (size_t)(c0 + 1) * CC * DIMD + wgBase,
                      vtOff0 + (unsigned)(((c0 + 1) & 1) * CC * BCOL * 4));

    // ---- Phase A: per-channel decay scan; build r~, k~, P_C; u-dot ----
    {
      float dloc[CC];
      #pragma unroll
      for (int s = 0; s < CC; ++s) dloc[s] = 0.0f;
      for (int a = tid; a < DIMD; a += NTHR) {
        float p  = 1.0f;
        float uv = u[a];
        const int gbase = c0 * CC * DIMD + a;
        #pragma unroll
        for (int s = 0; s < CC; ++s) {
          int   gi = gbase + s * DIMD;
          float rv = r[gi];
          float kv = k[gi];
          rt[s * DIMD + a] = rv * p;                      // r~ uses exclusive P
          float wv = fmaxf(w[gi], 0.0078125f);            // clamp: 1/P <= 2^112
          p *= wv;
          kt[s * DIMD + a] = kv / p;                      // k~ = k / P_{s+1}
          dloc[s] += rv * uv * kv;                        // diagonal term r.(u*k)
        }
        Pt[a] = p;
      }
      #pragma unroll
      for (int s = 0; s < CC; ++s) atomicAdd(&dtmp[s], dloc[s]);  // ds_add_f32
    }
    __syncthreads();

    // ---- Phase B1: intra scores Sc = R~ K~^T  (16x16, K=512), wave 0 ----
    if (wid == 0) {
      v8f a8 = {0.f, 0.f, 0.f, 0.f, 0.f, 0.f, 0.f, 0.f};
      #pragma unroll 8
      for (int kb = 0; kb < DIMD / 4; ++kb) {
        int a0k = kb * 4 + 2 * hi;
        v2f af, bf;
        af.x = rt[ln * DIMD + a0k];  af.y = rt[ln * DIMD + a0k + 1]; // A: M=tau,K=a
        bf.x = kt[ln * DIMD + a0k];  bf.y = kt[ln * DIMD + a0k + 1]; // B: K=a,N=s
        a8 = __builtin_amdgcn_wmma_f32_16x16x4_f32(false, af, false, bf,
                                                   (short)0, a8, false, false);
      }
      #pragma unroll
      for (int j = 0; j < 8; ++j) sc[(j + 8 * hi) * CC + ln] = a8[j];
    }
    __syncthreads();

    // ---- Phase B2: causal mask (s>tau -> 0) and diagonal = r.(u*k) ----
    if (tid < CC) {
      int tau = tid;
      sc[tau * CC + tau] = dtmp[tau];
      for (int s = tau + 1; s < CC; ++s) sc[tau * CC + s] = 0.0f;
    }
    // TDM: in-order completion per wave => waiting to depth 1 guarantees the
    // *older* (this chunk's) tile has landed while next chunk's stays in flight.
    if (wid == 3) {
      if (c0 + 1 < NCHUNK) __builtin_amdgcn_s_wait_tensorcnt(1);
      else                 __builtin_amdgcn_s_wait_tensorcnt(0);
    }
    __syncthreads();

    // prefetch next chunk's scan inputs into caches (global_prefetch_b8)
    if (c0 + 1 < NCHUNK) {
      const int nb = (c0 + 1) * CC * DIMD + tid * 64;
      __builtin_prefetch(&r[nb], 0, 0);
      __builtin_prefetch(&w[nb], 0, 0);
      __builtin_prefetch(&k[nb], 0, 0);
    }

    const float* vb = vt + (c0 & 1) * (CC * BCOL);   // this chunk's v buffer

    // ---- Phase C: Y tile = R~ @ S0 (K=512) + Sc @ V (K=16), per-wave 16 cols ----
    {
      v8f y8 = {0.f, 0.f, 0.f, 0.f, 0.f, 0.f, 0.f, 0.f};
      #pragma unroll 8
      for (int kb = 0; kb < DIMD / 4; ++kb) {
        int a0k = kb * 4 + 2 * hi;
        v2f af, bf;
        af.x = rt[ln * DIMD + a0k];          af.y = rt[ln * DIMD + a0k + 1];
        bf.x = st[a0k * SP + n0 + ln];       bf.y = st[(a0k + 1) * SP + n0 + ln];
        y8 = __builtin_amdgcn_wmma_f32_16x16x4_f32(false, af, false, bf,
                                                   (short)0, y8, false, false);
      }
      #pragma unroll
      for (int kb = 0; kb < 4; ++kb) {
        int s0 = kb * 4 + 2 * hi;
        v2f af, bf;
        af.x = sc[ln * CC + s0];             af.y = sc[ln * CC + s0 + 1];
        bf.x = vb[s0 * BCOL + n0 + ln];      bf.y = vb[(s0 + 1) * BCOL + n0 + ln];
        y8 = __builtin_amdgcn_wmma_f32_16x16x4_f32(false, af, false, bf,
                                                   (short)0, y8, false, false);
      }
      #pragma unroll
      for (int j = 0; j < 8; ++j)
        yout[(size_t)(c0 * CC + j + 8 * hi) * DIMD + wgBase + n0 + ln] = y8[j];
    }

    // ---- Phase D: state update S = diag(P_C) S + K^hat^T V, own 16-col stripe ----
    for (int it = 0; it < DIMD / 16; ++it) {
      int   a0 = it * 16;
      float pa = Pt[a0 + ln];                 // decay for A-frag row a
      v8f c8;
      #pragma unroll
      for (int j = 0; j < 8; ++j) {
        int row = a0 + j + 8 * hi;
        c8[j] = st[row * SP + n0 + ln] * Pt[row];   // diag(P_C) * S
      }
      #pragma unroll
      for (int kb = 0; kb < 4; ++kb) {
        int s0 = kb * 4 + 2 * hi;
        v2f af, bf;
        af.x = kt[s0 * DIMD + a0 + ln] * pa;        // k^hat = k~ * P_C
        af.y = kt[(s0 + 1) * DIMD + a0 + ln] * pa;
        bf.x = vb[s0 * BCOL + n0 + ln];
        bf.y = vb[(s0 + 1) * BCOL + n0 + ln];
        c8 = __builtin_amdgcn_wmma_f32_16x16x4_f32(false, af, false, bf,
                                                   (short)0, c8, false, false);
      }
      #pragma unroll
      for (int j = 0; j < 8; ++j)
        st[(a0 + j + 8 * hi) * SP + n0 + ln] = c8[j];
    }
    __syncthreads();
  }

  // final state -> out[T*512 ..]
  for (int i = tid; i < DIMD * BCOL; i += NTHR) {
    int a = i >> 6, b = i & 63;
    sout[(size_t)a * DIMD + wgBase + b] = st[a * SP + b];
  }
}

extern "C" void kernel_launch(void* const* d_in, const int* in_sizes, int n_in,
                              void* d_out, int out_size, void* d_ws, size_t ws_size,
                              hipStream_t stream) {
  const float* r   = (const float*)d_in[0];
  const float* w   = (const float*)d_in[1];
  const float* k   = (const float*)d_in[2];
  const float* v   = (const float*)d_in[3];
  const float* is_ = (const float*)d_in[4];
  const float* u   = (const float*)d_in[5];
  float* out = (float*)d_out;

  const size_t shmem =
      (size_t)(DIMD * SP + 2 * CC * DIMD + 2 * CC * BCOL + DIMD + CC * CC + CC) *
      sizeof(float);  // ~216 KB, fits CDNA5 320KB/WGP
  (void)hipFuncSetAttribute((const void*)wkv_cdna5_kernel,
                            hipFuncAttributeMaxDynamicSharedMemorySize,
                            (int)shmem);
  wkv_cdna5_kernel<<<DIMD / BCOL, NTHR, shmem, stream>>>(r, w, k, v, is_, u, out);
}